// HGCN_pyg_31353261261173
// MI455X (gfx1250) — compile-verified
//
#include <hip/hip_runtime.h>
#include <hip/hip_bf16.h>

typedef float v2f __attribute__((ext_vector_type(2)));
typedef float v8f __attribute__((ext_vector_type(8)));

#define EPSF      1e-7f
#define MIN_NORMF 1e-15f

__device__ __forceinline__ float arcoshf_(float t) {
    return logf(t + sqrtf(fmaxf(t * t - 1.0f, MIN_NORMF)));
}

// ---------------------------------------------------------------- degrees ---
__global__ void k_deg_init(float* deg, int n) {
    int i = blockIdx.x * blockDim.x + threadIdx.x;
    if (i < n) deg[i] = 1.0f;                      // self loop
}
__global__ void k_deg_edge(const long long* __restrict__ dst, float* deg, long long E_) {
    long long e = (long long)blockIdx.x * blockDim.x + threadIdx.x;
    if (e < E_) atomicAdd(&deg[(int)dst[e]], 1.0f);
}
__global__ void k_deg_fin(float* deg, int n) {
    int i = blockIdx.x * blockDim.x + threadIdx.x;
    if (i < n) deg[i] = rsqrtf(fmaxf(deg[i], 1.0f));   // now holds dinv
}

// ------------------------------------------- layer1: logmap0 + GEMM (WMMA) ---
// h1[m][n] = sum_k logmap0(x)[m][k] * W1[n][k] + b1[n]   (M=16/wave, N=128, K=128)
__global__ __launch_bounds__(256) void k_gemm1(const float* __restrict__ x,
                                               const float* __restrict__ W1,
                                               const float* __restrict__ b1,
                                               float* __restrict__ h1, int n) {
    __shared__ float lds[8 * 16 * 128];
    const int lane = threadIdx.x & 31;
    const int wv   = threadIdx.x >> 5;
    const int tile = blockIdx.x * 8 + wv;
    if (tile * 16 >= n) return;                  // wave-uniform guard (EXEC stays full)
    const int row0 = tile * 16;
    float* sm = &lds[wv * (16 * 128)];

    // stage raw x tile (16 x 128) into LDS
    for (int r = 0; r < 16; ++r) {
        float4 g = *reinterpret_cast<const float4*>(&x[(size_t)(row0 + r) * 128 + lane * 4]);
        *reinterpret_cast<float4*>(&sm[r * 128 + lane * 4]) = g;
    }
    asm volatile("" ::: "memory");               // LDS is in-order per wave

    // logmap0 per row: lane handles row (lane&15), half (lane>>4) of the columns
    const int rrow = lane & 15;
    const int half = lane >> 4;
    float ss = 0.0f;
    for (int i = 0; i < 16; ++i) {
        const int c = half * 64 + i * 4;
        float4 v = *reinterpret_cast<float4*>(&sm[rrow * 128 + c]);
        if (c == 0) v.x = 0.0f;                  // exclude time coordinate
        ss += v.x * v.x + v.y * v.y + v.z * v.z + v.w * v.w;
    }
    ss += __shfl_xor(ss, 16, 32);
    const float yn    = fmaxf(sqrtf(ss), MIN_NORMF);
    const float x0    = sm[rrow * 128];
    const float theta = fmaxf(x0, 1.0f + EPSF);
    const float scale = arcoshf_(theta) / yn;
    asm volatile("" ::: "memory");
    for (int i = 0; i < 16; ++i) {
        const int c = half * 64 + i * 4;
        float4 v = *reinterpret_cast<float4*>(&sm[rrow * 128 + c]);
        v.x *= scale; v.y *= scale; v.z *= scale; v.w *= scale;
        if (c == 0) v.x = 0.0f;                  // logmap0 zeroes column 0
        *reinterpret_cast<float4*>(&sm[rrow * 128 + c]) = v;
    }
    asm volatile("" ::: "memory");

    // WMMA f32 16x16x4: 32 K-steps, 8 N-tiles. Bias-initialized accumulators.
    const int nl = lane & 15;
    v8f acc[8];
#pragma unroll
    for (int nt = 0; nt < 8; ++nt) {
        const float bb = b1[nt * 16 + nl];
#pragma unroll
        for (int j = 0; j < 8; ++j) acc[nt][j] = bb;
    }
    const int koff = 2 * half;
    for (int kk = 0; kk < 32; ++kk) {
        const int kb = kk * 4 + koff;
        const v2f a = *reinterpret_cast<const v2f*>(&sm[rrow * 128 + kb]);
#pragma unroll
        for (int nt = 0; nt < 8; ++nt) {
            const int ncol = nt * 16 + nl;
            const v2f b = *reinterpret_cast<const v2f*>(&W1[(size_t)ncol * 128 + kb]);
            acc[nt] = __builtin_amdgcn_wmma_f32_16x16x4_f32(
                false, a, false, b, (short)0, acc[nt], false, false);
        }
    }
#pragma unroll
    for (int nt = 0; nt < 8; ++nt) {
#pragma unroll
        for (int j = 0; j < 8; ++j) {
            const int m = j + 8 * half;
            h1[(size_t)(row0 + m) * 128 + nt * 16 + nl] = acc[nt][j];
        }
    }
}

// ------------------------------------------------------- graph aggregation ---
__global__ void k_agg1_init(const float* __restrict__ h1, const float* __restrict__ dinv,
                            float* __restrict__ agg1, int n) {
    int idx = blockIdx.x * blockDim.x + threadIdx.x;
    if (idx < n * 128) {
        int i = idx >> 7;
        float di = dinv[i];
        agg1[idx] = h1[idx] * di * di;           // self-loop term
    }
}
__global__ __launch_bounds__(256) void k_edge1(const long long* __restrict__ src,
                                               const long long* __restrict__ dst,
                                               const float* __restrict__ dinv,
                                               const float* __restrict__ h1,
                                               float* __restrict__ agg1, long long E_) {
    long long e = (long long)blockIdx.x * 8 + (threadIdx.x >> 5);
    int lane = threadIdx.x & 31;
    if (e >= E_) return;
    int s = (int)src[e], d = (int)dst[e];
    float nrm = dinv[s] * dinv[d];
    float4 v = *reinterpret_cast<const float4*>(&h1[(size_t)s * 128 + lane * 4]);
    float* o = &agg1[(size_t)d * 128 + lane * 4];
    atomicAdd(o + 0, v.x * nrm);
    atomicAdd(o + 1, v.y * nrm);
    atomicAdd(o + 2, v.z * nrm);
    atomicAdd(o + 3, v.w * nrm);
}

// --------------- fused: proj_tan0 -> expmap0 -> proj -> hyp_act -> logmap0 ---
__global__ __launch_bounds__(256) void k_act(const float* __restrict__ agg1,
                                             float* __restrict__ xt2, int n) {
    int node = blockIdx.x * 8 + (threadIdx.x >> 5);
    int lane = threadIdx.x & 31;
    if (node >= n) return;
    float4 u = *reinterpret_cast<const float4*>(&agg1[(size_t)node * 128 + lane * 4]);
    if (lane == 0) u.x = 0.0f;                           // proj_tan0
    float ss = u.x * u.x + u.y * u.y + u.z * u.z + u.w * u.w;
#pragma unroll
    for (int o = 16; o > 0; o >>= 1) ss += __shfl_xor(ss, o, 32);
    // expmap0 + proj + logmap0 + relu
    float xn  = fmaxf(sqrtf(ss), MIN_NORMF);
    float k1  = sinhf(xn) / xn;                          // s = k1 * u (spatial)
    float ss2 = ss * k1 * k1;                            // ||s||^2
    float tt  = sqrtf(fmaxf(1.0f + ss2, EPSF));          // proj time coord
    float th  = fmaxf(tt, 1.0f + EPSF);
    float yn  = fmaxf(sqrtf(ss2), MIN_NORMF);
    float scl = arcoshf_(th) / yn * k1;
    float4 v;
    v.x = fmaxf(scl * u.x, 0.0f); v.y = fmaxf(scl * u.y, 0.0f);
    v.z = fmaxf(scl * u.z, 0.0f); v.w = fmaxf(scl * u.w, 0.0f);
    if (lane == 0) v.x = 0.0f;                           // proj_tan0
    // expmap0 + proj + logmap0 (layer-2 input tangent vector)
    float ss3 = v.x * v.x + v.y * v.y + v.z * v.z + v.w * v.w;
#pragma unroll
    for (int o = 16; o > 0; o >>= 1) ss3 += __shfl_xor(ss3, o, 32);
    float vn   = fmaxf(sqrtf(ss3), MIN_NORMF);
    float k2   = sinhf(vn) / vn;
    float ss4  = ss3 * k2 * k2;
    float tt2  = sqrtf(fmaxf(1.0f + ss4, EPSF));
    float th2  = fmaxf(tt2, 1.0f + EPSF);
    float yn2  = fmaxf(sqrtf(ss4), MIN_NORMF);
    float scl2 = arcoshf_(th2) / yn2 * k2;
    float4 w;
    w.x = scl2 * v.x; w.y = scl2 * v.y; w.z = scl2 * v.z; w.w = scl2 * v.w;
    if (lane == 0) w.x = 0.0f;
    *reinterpret_cast<float4*>(&xt2[(size_t)node * 128 + lane * 4]) = w;
}

// ------------------------------------------------- layer2 GEMM (WMMA, N=40) ---
__global__ __launch_bounds__(256) void k_gemm2(const float* __restrict__ xt2,
                                               const float* __restrict__ W2,
                                               const float* __restrict__ b2,
                                               float* __restrict__ h2, int n) {
    const int lane = threadIdx.x & 31;
    const int wv   = threadIdx.x >> 5;
    const int tile = blockIdx.x * 8 + wv;
    if (tile * 16 >= n) return;
    const int row0 = tile * 16;
    const int rrow = lane & 15;
    const int half = lane >> 4;
    const int nl   = lane & 15;
    v8f acc[3];
#pragma unroll
    for (int nt = 0; nt < 3; ++nt) {
        const int ncol = nt * 16 + nl;
        const float bb = (ncol < 40) ? b2[ncol] : 0.0f;
#pragma unroll
        for (int j = 0; j < 8; ++j) acc[nt][j] = bb;
    }
    const int koff = 2 * half;
    for (int kk = 0; kk < 32; ++kk) {
        const int kb = kk * 4 + koff;
        const v2f a = *reinterpret_cast<const v2f*>(&xt2[(size_t)(row0 + rrow) * 128 + kb]);
#pragma unroll
        for (int nt = 0; nt < 3; ++nt) {
            const int ncol = nt * 16 + nl;
            v2f b;
            if (ncol < 40) b = *reinterpret_cast<const v2f*>(&W2[(size_t)ncol * 128 + kb]);
            else { b.x = 0.0f; b.y = 0.0f; }
            acc[nt] = __builtin_amdgcn_wmma_f32_16x16x4_f32(
                false, a, false, b, (short)0, acc[nt], false, false);
        }
    }
#pragma unroll
    for (int nt = 0; nt < 3; ++nt) {
        const int ncol = nt * 16 + nl;
        if (ncol < 40) {
#pragma unroll
            for (int j = 0; j < 8; ++j) {
                const int m = j + 8 * half;
                h2[(size_t)(row0 + m) * 40 + ncol] = acc[nt][j];
            }
        }
    }
}

__global__ void k_agg2_init(const float* __restrict__ h2, const float* __restrict__ dinv,
                            float* __restrict__ agg2, int n) {
    int idx = blockIdx.x * blockDim.x + threadIdx.x;
    if (idx < n * 40) {
        int i = idx / 40;
        float di = dinv[i];
        agg2[idx] = h2[idx] * di * di;
    }
}
__global__ __launch_bounds__(320) void k_edge2(const long long* __restrict__ src,
                                               const long long* __restrict__ dst,
                                               const float* __restrict__ dinv,
                                               const float* __restrict__ h2,
                                               float* __restrict__ agg2, long long E_) {
    int t = threadIdx.x;
    long long e = (long long)blockIdx.x * 8 + t / 40;
    int col = t % 40;
    if (e >= E_) return;
    int s = (int)src[e], d = (int)dst[e];
    float nrm = dinv[s] * dinv[d];
    atomicAdd(&agg2[(size_t)d * 40 + col], h2[(size_t)s * 40 + col] * nrm);
}

// ------------- final: proj_tan0 -> expmap0 -> proj -> logmap0 -> log_softmax ---
__global__ __launch_bounds__(256) void k_out(const float* __restrict__ agg2,
                                             float* __restrict__ out, int n) {
    int node = blockIdx.x * 8 + (threadIdx.x >> 5);
    int lane = threadIdx.x & 31;
    if (node >= n) return;
    const float* row = &agg2[(size_t)node * 40];
    float u0 = row[lane];
    float u1 = (lane < 8) ? row[32 + lane] : 0.0f;
    if (lane == 0) u0 = 0.0f;                            // proj_tan0
    float ss = u0 * u0 + u1 * u1;
#pragma unroll
    for (int o = 16; o > 0; o >>= 1) ss += __shfl_xor(ss, o, 32);
    float xn  = fmaxf(sqrtf(ss), MIN_NORMF);
    float k1  = sinhf(xn) / xn;
    float ss2 = ss * k1 * k1;
    float tt  = sqrtf(fmaxf(1.0f + ss2, EPSF));
    float th  = fmaxf(tt, 1.0f + EPSF);
    float yn  = fmaxf(sqrtf(ss2), MIN_NORMF);
    float scl = arcoshf_(th) / yn * k1;
    float o0 = (lane == 0) ? 0.0f : scl * u0;
    float o1 = scl * u1;
    // log_softmax over the 40 logits
    float m = (lane < 8) ? fmaxf(o0, o1) : o0;
#pragma unroll
    for (int o = 16; o > 0; o >>= 1) m = fmaxf(m, __shfl_xor(m, o, 32));
    float esum = expf(o0 - m) + ((lane < 8) ? expf(o1 - m) : 0.0f);
#pragma unroll
    for (int o = 16; o > 0; o >>= 1) esum += __shfl_xor(esum, o, 32);
    float lse = logf(esum);
    out[(size_t)node * 40 + lane] = o0 - m - lse;
    if (lane < 8) out[(size_t)node * 40 + 32 + lane] = o1 - m - lse;
}

// ---------------------------------------------------------------- launcher ---
extern "C" void kernel_launch(void* const* d_in, const int* in_sizes, int n_in,
                              void* d_out, int out_size, void* d_ws, size_t ws_size,
                              hipStream_t stream) {
    const float*     x   = (const float*)d_in[0];
    const long long* ei  = (const long long*)d_in[1];   // int64 edge_index (2,E)
    const float*     W1  = (const float*)d_in[2];
    const float*     b1  = (const float*)d_in[3];
    const float*     W2  = (const float*)d_in[4];
    const float*     b2  = (const float*)d_in[5];
    float*           out = (float*)d_out;

    const int       n  = in_sizes[0] / 128;             // 100000
    const long long E_ = (long long)in_sizes[1] / 2;    // 1600000
    const long long* src = ei;
    const long long* dst = ei + E_;

    char*  ws   = (char*)d_ws;
    float* dinv = (float*)ws;
    size_t offA = ((size_t)n * 4 + 255) & ~(size_t)255;
    float* bufA = (float*)(ws + offA);                         // h1 / xt2 / agg2
    float* bufB = (float*)(ws + offA + (size_t)n * 128 * 4);   // agg1 / h2

    const int ntiles = (n + 15) / 16;
    const int gTile  = (ntiles + 7) / 8;
    const int gNode  = (n + 7) / 8;
    const int gEdge  = (int)((E_ + 7) / 8);

    k_deg_init<<<(n + 255) / 256, 256, 0, stream>>>(dinv, n);
    k_deg_edge<<<(int)((E_ + 255) / 256), 256, 0, stream>>>(dst, dinv, E_);
    k_deg_fin<<<(n + 255) / 256, 256, 0, stream>>>(dinv, n);

    k_gemm1<<<gTile, 256, 0, stream>>>(x, W1, b1, bufA, n);
    k_agg1_init<<<(n * 128 + 255) / 256, 256, 0, stream>>>(bufA, dinv, bufB, n);
    k_edge1<<<gEdge, 256, 0, stream>>>(src, dst, dinv, bufA, bufB, E_);
    k_act<<<gNode, 256, 0, stream>>>(bufB, bufA, n);

    k_gemm2<<<gTile, 256, 0, stream>>>(bufA, W2, b2, bufB, n);
    k_agg2_init<<<(n * 40 + 255) / 256, 256, 0, stream>>>(bufB, dinv, bufA, n);
    k_edge2<<<gEdge, 320, 0, stream>>>(src, dst, dinv, bufB, bufA, E_);
    k_out<<<gNode, 256, 0, stream>>>(bufA, out, n);
}